// LayerNorm_2104533975136
// MI455X (gfx1250) — compile-verified
//
#include <hip/hip_runtime.h>

typedef __attribute__((ext_vector_type(2))) float v2f;
typedef __attribute__((ext_vector_type(8))) float v8f;

#define NSEG 2048
#define SEG_EPS 1e-6f

// ---------------------------------------------------------------------------
// Zero the segment accumulators (4 * NSEG floats at start of workspace).
// ---------------------------------------------------------------------------
__global__ void zero_ws(float* __restrict__ ws, int n) {
    int i = blockIdx.x * blockDim.x + threadIdx.x;
    if (i < n) ws[i] = 0.0f;
}

// ---------------------------------------------------------------------------
// Per-node statistics via V_WMMA_F32_16X16X4_F32 with an all-ones B matrix.
// One wave handles a 16-node tile. A-matrix f32 16x4 layout (ISA 7.12.2):
//   lanes 0-15 : row M = lane,    VGPR0=K0, VGPR1=K1  -> contiguous float2
//   lanes 16-31: row M = lane-16, VGPR0=K2, VGPR1=K3  -> contiguous float2
// With B == 1, D[m][n] = row-sum(m) for every n. C/D layout: lane 0 holds
// rows 0..7 in its 8 accumulator VGPRs, lane 16 holds rows 8..15.
// Three accumulator chains: sum(s), sum(s^2), sum(v^2).
// ---------------------------------------------------------------------------
__global__ __launch_bounds__(128)
void node_reduce_wmma(const float* __restrict__ s, const float* __restrict__ v,
                      const int* __restrict__ batch,
                      float* __restrict__ segSum, float* __restrict__ segSq,
                      float* __restrict__ segV,   float* __restrict__ segCnt,
                      int tiles) {
    const int lane = threadIdx.x & 31;
    const int wave = threadIdx.x >> 5;
    const int tile = blockIdx.x * 4 + wave;
    if (tile >= tiles) return;                 // wave-uniform: EXEC stays all-1s

    const int nodeBase = tile * 16;
    const int row      = nodeBase + (lane & 15);
    const int khalf    = lane >> 4;            // 0 for lanes 0-15, 1 for 16-31

    const v2f bones = {1.0f, 1.0f};
    v8f accS = {};   // sum of s       over 256 channels per row
    v8f accQ = {};   // sum of s^2     over 256 channels per row
    v8f accV = {};   // sum of v^2     over 384 elements per row

    // ---- scalar branch: 256 channels = 64 K-steps of 4 ----
    const float* ps = s + (size_t)row * 256 + 2 * khalf;
#pragma unroll 4
    for (int k = 0; k < 64; ++k) {
        v2f a = *(const v2f*)ps; ps += 4;
        accS = __builtin_amdgcn_wmma_f32_16x16x4_f32(
                   false, a,  false, bones, (short)0, accS, false, false);
        v2f a2 = a * a;
        accQ = __builtin_amdgcn_wmma_f32_16x16x4_f32(
                   false, a2, false, bones, (short)0, accQ, false, false);
    }

    // ---- vector branch: 3*128 = 384 contiguous floats = 96 K-steps of 4 ----
    const float* pv = v + (size_t)row * 384 + 2 * khalf;
#pragma unroll 4
    for (int k = 0; k < 96; ++k) {
        v2f a  = *(const v2f*)pv; pv += 4;
        v2f a2 = a * a;
        accV = __builtin_amdgcn_wmma_f32_16x16x4_f32(
                   false, a2, false, bones, (short)0, accV, false, false);
    }

    // ---- lane 0 owns rows 0..7, lane 16 owns rows 8..15 ----
    if ((lane & 15) == 0) {
        const int rbase = khalf * 8;
#pragma unroll
        for (int r = 0; r < 8; ++r) {
            const int node = nodeBase + rbase + r;
            const int g    = batch[node];
            atomicAdd(&segSum[g], accS[r]);
            atomicAdd(&segSq[g],  accQ[r]);
            atomicAdd(&segV[g],   accV[r]);
            atomicAdd(&segCnt[g], 1.0f);
        }
    }
}

// Scalar tail for N % 16 != 0 (not taken for N = 200000; kept for generality).
__global__ void node_reduce_tail(const float* __restrict__ s, const float* __restrict__ v,
                                 const int* __restrict__ batch,
                                 float* segSum, float* segSq, float* segV, float* segCnt,
                                 int start, int nrem) {
    int i = blockIdx.x * blockDim.x + threadIdx.x;
    if (i >= nrem) return;
    int node = start + i;
    const float* ps = s + (size_t)node * 256;
    float su = 0.f, sq = 0.f;
    for (int c = 0; c < 256; ++c) { float x = ps[c]; su += x; sq += x * x; }
    const float* pv = v + (size_t)node * 384;
    float vs = 0.f;
    for (int c = 0; c < 384; ++c) { float x = pv[c]; vs += x * x; }
    int g = batch[node];
    atomicAdd(&segSum[g], su);
    atomicAdd(&segSq[g],  sq);
    atomicAdd(&segV[g],   vs);
    atomicAdd(&segCnt[g], 1.0f);
}

// ---------------------------------------------------------------------------
// Per-segment finalize:  smean, 1/var (reference divides by var, not sqrt),
// and 1/vmean.  var = E[mean_c s^2] - smean^2  (single-pass identity).
// ---------------------------------------------------------------------------
__global__ void seg_finalize(const float* __restrict__ segSum, const float* __restrict__ segSq,
                             const float* __restrict__ segV,   const float* __restrict__ segCnt,
                             float* __restrict__ smean, float* __restrict__ invvar,
                             float* __restrict__ invv) {
    int g = blockIdx.x * blockDim.x + threadIdx.x;
    if (g >= NSEG) return;
    float cnt = fmaxf(segCnt[g], 1.0f);
    float m   = segSum[g] / (256.0f * cnt);
    float ex2 = segSq[g]  / (256.0f * cnt);
    float var = fmaxf(ex2 - m * m, SEG_EPS);
    float vm  = fmaxf(segV[g] / (128.0f * cnt), SEG_EPS);
    smean[g]  = m;
    invvar[g] = 1.0f / var;
    invv[g]   = 1.0f / vm;
}

// ---------------------------------------------------------------------------
// sout = (s - smean[g]) * invvar[g] * weight + bias   (float4 streaming)
// one thread per float4; 64 float4 per node row.
// ---------------------------------------------------------------------------
__global__ __launch_bounds__(256)
void sout_kernel(const float* __restrict__ s, const int* __restrict__ batch,
                 const float* __restrict__ wgt, const float* __restrict__ bias,
                 const float* __restrict__ smean, const float* __restrict__ invvar,
                 float* __restrict__ out, int n4) {
    int idx = blockIdx.x * blockDim.x + threadIdx.x;
    if (idx >= n4) return;
    int n  = idx >> 6;
    int c4 = idx & 63;
    int g  = batch[n];
    float4 sv = ((const float4*)s)[idx];
    float4 wv = ((const float4*)wgt)[c4];
    float4 bv = ((const float4*)bias)[c4];
    float  m  = smean[g];
    float  iv = invvar[g];
    float4 o;
    o.x = (sv.x - m) * iv * wv.x + bv.x;
    o.y = (sv.y - m) * iv * wv.y + bv.y;
    o.z = (sv.z - m) * iv * wv.z + bv.z;
    o.w = (sv.w - m) * iv * wv.w + bv.w;
    ((float4*)out)[idx] = o;
}

// ---------------------------------------------------------------------------
// vout = v * invvmean[g]   (float4 streaming); 96 float4 per node row.
// ---------------------------------------------------------------------------
__global__ __launch_bounds__(256)
void vout_kernel(const float* __restrict__ v, const int* __restrict__ batch,
                 const float* __restrict__ invv, float* __restrict__ out, int n4) {
    int idx = blockIdx.x * blockDim.x + threadIdx.x;
    if (idx >= n4) return;
    int n = idx / 96;
    float sc = invv[batch[n]];
    float4 vv = ((const float4*)v)[idx];
    float4 o = { vv.x * sc, vv.y * sc, vv.z * sc, vv.w * sc };
    ((float4*)out)[idx] = o;
}

// ---------------------------------------------------------------------------
extern "C" void kernel_launch(void* const* d_in, const int* in_sizes, int n_in,
                              void* d_out, int out_size, void* d_ws, size_t ws_size,
                              hipStream_t stream) {
    const float* s     = (const float*)d_in[0];   // [N,256]
    const float* v     = (const float*)d_in[1];   // [N,3,128]
    const float* wgt   = (const float*)d_in[2];   // [256]
    const float* bias  = (const float*)d_in[3];   // [256]
    const int*   batch = (const int*)d_in[4];     // [N]
    const int N = in_sizes[4];

    float* ws     = (float*)d_ws;
    float* segSum = ws + 0 * NSEG;
    float* segSq  = ws + 1 * NSEG;
    float* segV   = ws + 2 * NSEG;
    float* segCnt = ws + 3 * NSEG;
    float* smean  = ws + 4 * NSEG;
    float* invvar = ws + 5 * NSEG;
    float* invv   = ws + 6 * NSEG;
    float* out    = (float*)d_out;

    // 1) zero accumulators (deterministic per call)
    zero_ws<<<(4 * NSEG + 255) / 256, 256, 0, stream>>>(ws, 4 * NSEG);

    // 2) per-node stats -> segment atomics (WMMA row-sum tiles)
    const int tiles = N / 16;
    if (tiles > 0) {
        node_reduce_wmma<<<(tiles + 3) / 4, 128, 0, stream>>>(
            s, v, batch, segSum, segSq, segV, segCnt, tiles);
    }
    const int rem = N - tiles * 16;
    if (rem > 0) {
        node_reduce_tail<<<1, 32, 0, stream>>>(
            s, v, batch, segSum, segSq, segV, segCnt, tiles * 16, rem);
    }

    // 3) per-segment finalize
    seg_finalize<<<(NSEG + 255) / 256, 256, 0, stream>>>(
        segSum, segSq, segV, segCnt, smean, invvar, invv);

    // 4) elementwise outputs
    const int n4s = N * 64;   // N*256/4
    sout_kernel<<<(n4s + 255) / 256, 256, 0, stream>>>(
        s, batch, wgt, bias, smean, invvar, out, n4s);

    const int n4v = N * 96;   // N*384/4
    vout_kernel<<<(n4v + 255) / 256, 256, 0, stream>>>(
        v, batch, invv, out + (size_t)N * 256, n4v);
}